// Splatting_43980465111475
// MI455X (gfx1250) — compile-verified
//
#include <hip/hip_runtime.h>

// Reference shapes: frame [8,3,512,1024] f32, flow [8,2,512,1024] f32 -> out [8,3,512,1024] f32
constexpr int N_ = 8;
constexpr int C_ = 3;
constexpr int H_ = 512;
constexpr int W_ = 1024;

constexpr int TW = 64;          // tile width
constexpr int TH = 64;          // tile height
constexpr int HALO = 4;         // |flow| <= 4 captured in LDS (~99.99% of N(0,1))
constexpr int LW = TW + 2 * HALO;   // 72
constexpr int LH = TH + 2 * HALO;   // 72
constexpr int LTILE = LW * LH;      // 5184 floats per channel
constexpr int NTHREADS = 1024;      // 32 wave32s -> saturates one WGP per block

// Native L2 float atomic, device scope (cross-WGP visibility), non-returning.
__device__ __forceinline__ void global_fadd(float* p, float v) {
  asm volatile("global_atomic_add_f32 %0, %1, off scope:SCOPE_DEV"
               :: "v"(p), "v"(v) : "memory");
}

__global__ void zero_f4(float4* __restrict__ out, int n4) {
  int i = blockIdx.x * blockDim.x + threadIdx.x;
  if (i < n4) out[i] = make_float4(0.f, 0.f, 0.f, 0.f);
}

__global__ __launch_bounds__(NTHREADS)
void splat_kernel(const float* __restrict__ frame,
                  const float* __restrict__ flow,
                  float* __restrict__ out) {
  __shared__ __align__(16) float acc[C_ * LTILE];   // 62,208 B of 320 KB/WGP

  const int n   = blockIdx.z;
  const int tx0 = blockIdx.x * TW;
  const int ty0 = blockIdx.y * TH;
  const int tid = threadIdx.x;

  // Zero the private accumulator with b128 LDS stores.
  {
    float4* a4 = (float4*)acc;
    for (int i = tid; i < (C_ * LTILE) / 4; i += NTHREADS)
      a4[i] = make_float4(0.f, 0.f, 0.f, 0.f);
  }
  __syncthreads();

  const int hw = H_ * W_;
  const float* fx_p = flow  + (n * 2 + 0) * hw;
  const float* fy_p = flow  + (n * 2 + 1) * hw;
  const float* f0   = frame + (n * C_ + 0) * hw;
  const float* f1   = frame + (n * C_ + 1) * hw;
  const float* f2   = frame + (n * C_ + 2) * hw;
  float* o0 = out + (n * C_ + 0) * hw;
  float* o1 = out + (n * C_ + 1) * hw;
  float* o2 = out + (n * C_ + 2) * hw;

  for (int p = tid; p < TW * TH; p += NTHREADS) {
    const int lx  = p & (TW - 1);
    const int ly  = p >> 6;              // TW == 64
    const int x   = tx0 + lx;
    const int y   = ty0 + ly;
    const int pix = y * W_ + x;

    // Prefetch the next chunk of this tile (global_prefetch_b8).
    if (p + NTHREADS < TW * TH) {
      const int pn   = p + NTHREADS;
      const int pixn = (ty0 + (pn >> 6)) * W_ + (tx0 + (pn & (TW - 1)));
      __builtin_prefetch(&fx_p[pixn], 0, 3);
      __builtin_prefetch(&f0[pixn], 0, 3);
    }

    const float fx = fx_p[pix];
    const float fy = fy_p[pix];
    const float v0 = f0[pix];
    const float v1 = f1[pix];
    const float v2 = f2[pix];

    const float txf = (float)x + fx;
    const float tyf = (float)y + fy;
    const float x0f = floorf(txf);
    const float y0f = floorf(tyf);
    const int   x0  = (int)x0f;
    const int   y0  = (int)y0f;
    const float ax  = txf - x0f;       // in [0,1)
    const float ay  = tyf - y0f;
    const float bx  = 1.f - ax;
    const float by  = 1.f - ay;

    auto deposit = [&](int cx, int cy, float w) {
      const int llx = cx - (tx0 - HALO);
      const int lly = cy - (ty0 - HALO);
      if ((unsigned)llx < (unsigned)LW && (unsigned)lly < (unsigned)LH) {
        // In-tile (incl. halo): absorb contention in LDS (ds_add_f32).
        float* b = &acc[lly * LW + llx];
        atomicAdd(b + 0 * LTILE, v0 * w);
        atomicAdd(b + 1 * LTILE, v1 * w);
        atomicAdd(b + 2 * LTILE, v2 * w);
      } else if ((unsigned)cx < (unsigned)W_ && (unsigned)cy < (unsigned)H_) {
        // Rare far splat: straight to device-scope L2 atomic.
        const int o = cy * W_ + cx;
        global_fadd(o0 + o, v0 * w);
        global_fadd(o1 + o, v1 * w);
        global_fadd(o2 + o, v2 * w);
      }
      // Out of image and out of halo: dropped (matches reference validity mask).
    };

    deposit(x0,     y0,     bx * by);  // NW
    deposit(x0 + 1, y0,     ax * by);  // NE
    deposit(x0,     y0 + 1, bx * ay);  // SW
    deposit(x0 + 1, y0 + 1, ax * ay);  // SE
  }

  __syncthreads();

  // Flush accumulator. Must be atomic: adjacent tiles' halos overlap and far
  // fallback splats may land anywhere. Skip exact zeros.
  const bool interior = (tx0 >= HALO) && (ty0 >= HALO) &&
                        (tx0 + TW + HALO <= W_) && (ty0 + TH + HALO <= H_);
  if (interior) {
    // Fast path: every LDS cell maps to a valid image cell.
    for (int i = tid; i < LTILE; i += NTHREADS) {
      const int gx = tx0 - HALO + (i % LW);
      const int gy = ty0 - HALO + (i / LW);
      const int o  = gy * W_ + gx;
      const float a0 = acc[i];
      const float a1 = acc[i + LTILE];
      const float a2 = acc[i + 2 * LTILE];
      if (a0 != 0.f) global_fadd(o0 + o, a0);
      if (a1 != 0.f) global_fadd(o1 + o, a1);
      if (a2 != 0.f) global_fadd(o2 + o, a2);
    }
  } else {
    for (int i = tid; i < LTILE; i += NTHREADS) {
      const int gx = tx0 - HALO + (i % LW);
      const int gy = ty0 - HALO + (i / LW);
      if ((unsigned)gx < (unsigned)W_ && (unsigned)gy < (unsigned)H_) {
        const int o  = gy * W_ + gx;
        const float a0 = acc[i];
        const float a1 = acc[i + LTILE];
        const float a2 = acc[i + 2 * LTILE];
        if (a0 != 0.f) global_fadd(o0 + o, a0);
        if (a1 != 0.f) global_fadd(o1 + o, a1);
        if (a2 != 0.f) global_fadd(o2 + o, a2);
      }
    }
  }
}

extern "C" void kernel_launch(void* const* d_in, const int* in_sizes, int n_in,
                              void* d_out, int out_size, void* d_ws, size_t ws_size,
                              hipStream_t stream) {
  const float* frame = (const float*)d_in[0];
  const float* flow  = (const float*)d_in[1];
  float* out = (float*)d_out;

  // d_out is poisoned by the harness: zero it (vectorized float4 stores).
  const int n4 = out_size / 4;   // 12,582,912 / 4
  zero_f4<<<(n4 + 255) / 256, 256, 0, stream>>>((float4*)out, n4);

  dim3 grid(W_ / TW, H_ / TH, N_);   // (16, 8, 8) = 1024 tiles
  splat_kernel<<<grid, NTHREADS, 0, stream>>>(frame, flow, out);
}